// CapsuleLayer_9852654977072
// MI455X (gfx1250) — compile-verified
//
#include <hip/hip_runtime.h>
#include <math.h>

// CDNA5 (gfx1250) fused capsule layer:
//   dense 5x5 conv (implicit GEMM via V_WMMA_F32_16X16X4_F32)
//   + 1/(8*cnt) normalization + squash over z1.
//
// Shapes (fp32): u[4,4,16,128,128], W[4,128,16,5,5] -> out[4,8,16,128,128]
//
// Staging uses GLOBAL_LOAD_ASYNC_TO_LDS_B32 (ASYNCcnt) when the builtin is
// available: both staging stages are global->LDS transposes (contiguous
// global reads, strided per-lane LDS scatter), which is exactly the async
// path's per-lane VDST addressing model. Falls back to VGPR round-trip.

typedef __attribute__((ext_vector_type(2))) float v2f;
typedef __attribute__((ext_vector_type(8))) float v8f;

#define NC    4
#define T0C   4
#define Z0C   16
#define T1C   8
#define Z1C   16
#define KC    5
#define HC    128
#define WC    128
#define CIN   64          // T0C*Z0C
#define COUT  128         // T1C*Z1C
#define TAPS  25          // 5*5

// LDS padding for conflict-free wave32 ds_load_b64:
//  - weight row stride 1604 floats (1604 % 64 == 4 -> lane m at bank 4m)
//  - input cin stride 68 floats   (68 % 64 == 4  -> lane n at bank 4n)
#define WSTR  1604
#define IW    132          // 128 + 2*pad
#define CSTR  68
#define WLDS_FLOATS (16 * WSTR)          // 25,664 floats
#define ILDS_FLOATS (KC * IW * CSTR)     // 44,880 floats
// total dynamic LDS = 282,176 bytes (< 320 KB per WGP)

#if __has_builtin(__builtin_amdgcn_global_load_async_to_lds_b32)
#define ASYNC_STAGE 1
typedef __attribute__((address_space(1))) int gas_int;   // global (AS1)
typedef __attribute__((address_space(3))) int las_int;   // LDS (AS3)
#else
#define ASYNC_STAGE 0
#endif

__device__ __forceinline__ void stage_b32(const float* gsrc, float* ldst) {
#if ASYNC_STAGE
    // C-style casts: const-strip + bitcast + addrspacecast in one step
    // (generic->AS3 must be a true addrspacecast: flat addr -> LDS offset).
    __builtin_amdgcn_global_load_async_to_lds_b32(
        (gas_int*)gsrc, (las_int*)ldst, 0, 0);
#else
    *ldst = *gsrc;
#endif
}

__device__ __forceinline__ void wait_async_then_barrier() {
#if ASYNC_STAGE
#if __has_builtin(__builtin_amdgcn_s_wait_asynccnt)
    __builtin_amdgcn_s_wait_asynccnt(0);
#else
    asm volatile("s_wait_asynccnt 0" ::: "memory");
#endif
#endif
    __syncthreads();
}

__global__ __launch_bounds__(256)
void capsule_fused_kernel(const float* __restrict__ u,
                          const float* __restrict__ Wt,
                          float* __restrict__ out) {
    extern __shared__ float smem[];
    float* wlds = smem;                  // [16 couts][WSTR]  k = tap*64 + cin
    float* ilds = smem + WLDS_FLOATS;    // [5 rows][132 cols][CSTR]

    const int tid = threadIdx.x;
    const int bx  = blockIdx.x;
    const int ct  = bx & 7;            // cout tile == t1 index
    const int h   = (bx >> 3) & 127;   // output row
    const int n   = bx >> 10;          // batch

    // ---- Stage weight tile: wlds[m][tap*64 + cin] = W[t0][ct*16+m][z0][kh][kw]
    // Global reads contiguous in tap; LDS scatter at stride 64 floats.
    for (int i = tid; i < 16 * CIN * TAPS; i += 256) {
        const int t   = i % TAPS;
        const int cin = (i / TAPS) % CIN;
        const int m   = i / (TAPS * CIN);
        const int t0  = cin >> 4, z0 = cin & 15;
        const int cout = (ct << 4) + m;
        stage_b32(&Wt[(((t0 * COUT + cout) * Z0C + z0) * TAPS) + t],
                  &wlds[m * WSTR + t * CIN + cin]);
    }

    // ---- Stage input band rows h-2..h+2, cols -2..129 (zero padded), all cin.
    // Global reads contiguous in col; LDS scatter at stride CSTR floats.
    for (int i = tid; i < CIN * KC * IW; i += 256) {
        const int col = i % IW;
        const int r   = (i / IW) % KC;
        const int cin = i / (IW * KC);
        const int hh  = h - 2 + r;
        const int ww  = col - 2;
        float* dst = &ilds[(r * IW + col) * CSTR + cin];
        if ((unsigned)hh < 128u && (unsigned)ww < 128u) {
            const int t0 = cin >> 4, z0 = cin & 15;
            stage_b32(&u[(((n * T0C + t0) * Z0C + z0) * HC + hh) * WC + ww], dst);
        } else {
            *dst = 0.0f;   // halo zero-fill (DScnt path, disjoint addresses)
        }
    }

    // Speculative L2 prefetch of the next output row's new input line (h+3).
    if (h + 3 < HC) {
        __builtin_prefetch(u + ((size_t)(n * T0C) * Z0C * HC + (h + 3)) * WC, 0, 1);
    }

    wait_async_then_barrier();

    // ---- Per-wave 16(cout) x 16(w) tile, K = 1600 in steps of 4
    const int lane  = tid & 31;
    const int wave  = tid >> 5;
    const int w0    = wave << 4;       // 8 waves x 16 cols = 128 cols
    const int nl    = lane & 15;       // A: M index | B: N (w) index
    const int khalf = lane >> 4;       // K sub-pair: lanes 16-31 hold K+2,K+3

    v8f acc0 = {};
    v8f acc1 = {};

    const float* Arow = &wlds[nl * WSTR + khalf * 2];

    for (int t = 0; t < TAPS; ++t) {
        const int kh = t / 5, kw = t % 5;
        const float* A = Arow + t * CIN;
        const float* B = &ilds[((kh * IW) + (w0 + nl + kw)) * CSTR + khalf * 2];
        #pragma unroll
        for (int c4 = 0; c4 < 16; c4 += 2) {
            // A fragment 16x4: lane(m=nl) v[0..1] = W[m][k0 + 2*khalf + 0..1]
            // B fragment 4x16: lane(n=nl) v[0..1] = X[k0 + 2*khalf + 0..1][n]
            v2f a0 = *(const v2f*)(A + c4 * 4);
            v2f b0 = *(const v2f*)(B + c4 * 4);
            acc0 = __builtin_amdgcn_wmma_f32_16x16x4_f32(
                false, a0, false, b0, (short)0, acc0, false, false);
            v2f a1 = *(const v2f*)(A + (c4 + 1) * 4);
            v2f b1 = *(const v2f*)(B + (c4 + 1) * 4);
            acc1 = __builtin_amdgcn_wmma_f32_16x16x4_f32(
                false, a1, false, b1, (short)0, acc1, false, false);
        }
    }

    v8f p = acc0 + acc1;

    // ---- Degenerate routing: p = y / (8 * cnt(h,w))
    const int w  = w0 + nl;
    const int r0 = (h >= 2) ? h - 2 : 0;
    const int r1 = (h + 2 <= 127) ? h + 2 : 127;
    const int c0 = (w >= 2) ? w - 2 : 0;
    const int c1 = (w + 2 <= 127) ? w + 2 : 127;
    const float invd = 1.0f / (8.0f * (float)((r1 - r0 + 1) * (c1 - c0 + 1)));

    float ss = 0.0f;
    #pragma unroll
    for (int r = 0; r < 8; ++r) {
        const float pv = p[r] * invd;
        p[r] = pv;
        ss += pv * pv;
    }
    // Column n lives in lanes {n, n+16} (M=0..7 / M=8..15): pairwise reduce.
    ss += __shfl_xor(ss, 16, 32);
    const float n2 = ss;
    const float scale = n2 / ((1.0f + n2) * sqrtf(n2 + 1e-9f));

    // ---- Store: out[n][t1=ct][z1][h][w], z1 = r + 8*khalf
    const size_t obase =
        ((((size_t)n * T1C + ct) * Z1C + (size_t)(khalf * 8)) * HC + h) * WC + w;
    #pragma unroll
    for (int r = 0; r < 8; ++r) {
        out[obase + (size_t)r * (HC * WC)] = p[r] * scale;
    }
}

extern "C" void kernel_launch(void* const* d_in, const int* in_sizes, int n_in,
                              void* d_out, int out_size, void* d_ws, size_t ws_size,
                              hipStream_t stream) {
    const float* u  = (const float*)d_in[0];   // [4,4,16,128,128] f32
    const float* Wt = (const float*)d_in[1];   // [4,128,16,5,5]  f32
    float* out = (float*)d_out;                // [4,8,16,128,128] f32

    const size_t shmem = (size_t)(WLDS_FLOATS + ILDS_FLOATS) * sizeof(float);
    (void)hipFuncSetAttribute((const void*)capsule_fused_kernel,
                              hipFuncAttributeMaxDynamicSharedMemorySize,
                              (int)shmem);

    dim3 grid(NC * HC * T1C);   // 4096 workgroups: (n, h, cout-tile)
    dim3 block(256);            // 8 waves (wave32)
    capsule_fused_kernel<<<grid, block, shmem, stream>>>(u, Wt, out);
}